// HDDTBinaryLoss_37726992728905
// MI455X (gfx1250) — compile-verified
//
#include <hip/hip_runtime.h>
#include <hip/hip_bf16.h>

// HDDT binary loss for MI455X (gfx1250, wave32).
// B=8, C=1, H=W=256. Output: single f32 scalar.
//
// K1: vertical 1D distance scans (fwd/bwd, BIG=512) -> g2 = min(fwd,bwd)^2, 4 mask planes.
// K2: row-wise exact lower envelope dt2[j] = min_k (j-k)^2 + g2[k] for all 4 planes,
//     staged into LDS via gfx1250 async load-to-LDS (ASYNCcnt path); fused
//     (t - sigmoid(x))^2 * dist; deterministic block tree-reduction to one partial per block.
// K3: single-wave v_wmma_f32_16x16x4_f32 ones-matmul reduction of the 2048 partials
//     (layout-agnostic: sum of all D entries == 16 * sum of B entries), /16, * 1/(B*C*H*W).

#define BB 8
#define HH 256
#define WW 256

typedef __attribute__((ext_vector_type(2))) float v2f;
typedef __attribute__((ext_vector_type(8))) float v8f;

typedef __attribute__((address_space(1))) int as1_int;
typedef __attribute__((address_space(3))) int as3_int;

#if defined(__gfx1250__) && __has_builtin(__builtin_amdgcn_global_load_async_to_lds_b32)
#define USE_ASYNC_LDS 1
#else
#define USE_ASYNC_LDS 0
#endif

__device__ __forceinline__ bool mask_bit(int m, float x, int t) {
    // m=0: sigmoid(x)>0.5 <=> x>0 ; m=1: complement ; m=2: t>0 ; m=3: complement
    if (m == 0) return x > 0.0f;
    if (m == 1) return !(x > 0.0f);
    if (m == 2) return t > 0;
    return !(t > 0);
}

// One block per (mask, b) plane; one thread per column j. Scans along rows.
__global__ void edt_pass1(const float* __restrict__ inp, const int* __restrict__ tgt,
                          float* __restrict__ g2) {
    const int plane = blockIdx.x;        // mask*BB + b
    const int m = plane >> 3;
    const int b = plane & 7;
    const int j = threadIdx.x;           // 0..WW-1
    const float BIG = (float)(HH + WW);  // 512, acts as +inf for the 1D pass

    const int baseIn = b * (HH * WW) + j;
    float* gp = g2 + plane * (HH * WW) + j;

    // forward scan: d = mask ? d+1 : 0, carry init BIG (matches jax.lax.scan init)
    float d = BIG;
    for (int i = 0; i < HH; ++i) {
        const int off = i * WW;
        bool bit = (m < 2) ? mask_bit(m, inp[baseIn + off], 0)
                           : mask_bit(m, 0.0f, tgt[baseIn + off]);
        d = bit ? (d + 1.0f) : 0.0f;
        gp[off] = d;
    }
    // backward scan, then square of the min of the two directions
    d = BIG;
    for (int i = HH - 1; i >= 0; --i) {
        const int off = i * WW;
        bool bit = (m < 2) ? mask_bit(m, inp[baseIn + off], 0)
                           : mask_bit(m, 0.0f, tgt[baseIn + off]);
        d = bit ? (d + 1.0f) : 0.0f;
        float mn = fminf(gp[off], d);
        gp[off] = mn * mn;
    }
}

// One block per (b, row i); 256 threads (one per output column j).
__global__ void edt_pass2_loss(const float* __restrict__ inp, const int* __restrict__ tgt,
                               const float* __restrict__ g2, float* __restrict__ partials) {
    __shared__ float lds[4][WW];
    __shared__ float red[WW];

    const int row = blockIdx.x;   // b*HH + i
    const int b = row >> 8;
    const int i = row & 255;
    const int j = threadIdx.x;

    // Stage the four g2 rows for this (b,i) into LDS.
#if USE_ASYNC_LDS
    #pragma unroll
    for (int m = 0; m < 4; ++m) {
        const float* src = g2 + (((m * BB + b) * HH + i) * WW) + j;
        __builtin_amdgcn_global_load_async_to_lds_b32(
            (as1_int*)src,
            (as3_int*)&lds[m][j],
            0, 0);
    }
#if __has_builtin(__builtin_amdgcn_s_wait_asynccnt)
    __builtin_amdgcn_s_wait_asynccnt(0);
#else
    asm volatile("s_wait_asynccnt 0" ::: "memory");
#endif
#else
    #pragma unroll
    for (int m = 0; m < 4; ++m)
        lds[m][j] = g2[(((m * BB + b) * HH + i) * WW) + j];
#endif
    __syncthreads();

    // Exact lower envelope: dt2_m = min_k (j-k)^2 + g2_m[k]; LDS reads are broadcasts.
    const float jf = (float)j;
    float b0 = 3.0e38f, b1 = 3.0e38f, b2 = 3.0e38f, b3 = 3.0e38f;
    for (int k = 0; k < WW; ++k) {
        float dk = jf - (float)k;
        float dd = dk * dk;                 // exact: |dk|<=255, squares < 2^24
        b0 = fminf(b0, dd + lds[0][k]);
        b1 = fminf(b1, dd + lds[1][k]);
        b2 = fminf(b2, dd + lds[2][k]);
        b3 = fminf(b3, dd + lds[3][k]);
    }
    const float dist = (b0 + b1) + (b2 + b3);

    const int idx = (b * HH + i) * WW + j;
    const float x = inp[idx];
    const float t = (float)tgt[idx];
    const float p = 1.0f / (1.0f + expf(-x));
    float pe = t - p;
    pe *= pe;

    // Deterministic block reduction (fixed tree order).
    red[j] = pe * dist;
    __syncthreads();
    #pragma unroll
    for (int s = 128; s > 0; s >>= 1) {
        if (j < s) red[j] += red[j + s];
        __syncthreads();
    }
    if (j == 0) partials[blockIdx.x] = red[0];
}

// Single wave32. A = ones(16x4); each iteration folds 64 partials into C.
// sum(all 256 D entries) == 16 * sum(all 64 B entries), independent of VGPR layout.
__global__ void wmma_reduce(const float* __restrict__ partials, float* __restrict__ out,
                            int n, float scale) {
    const int lane = threadIdx.x;  // 0..31, EXEC all ones for the WMMA loop
    v8f c = {};
    v2f a;
    a.x = 1.0f;
    a.y = 1.0f;
    for (int base = 0; base < n; base += 64) {
        v2f bv;
        bv.x = partials[base + lane * 2];
        bv.y = partials[base + lane * 2 + 1];
        // (neg_a, A, neg_b, B, c_mod, C, reuse_a, reuse_b)
        c = __builtin_amdgcn_wmma_f32_16x16x4_f32(false, a, false, bv,
                                                  (short)0, c, false, false);
    }
    float s = c[0] + c[1] + c[2] + c[3] + c[4] + c[5] + c[6] + c[7];
    #pragma unroll
    for (int off = 16; off > 0; off >>= 1) s += __shfl_xor(s, off, 32);
    if (lane == 0) out[0] = s * (1.0f / 16.0f) * scale;
}

extern "C" void kernel_launch(void* const* d_in, const int* in_sizes, int n_in,
                              void* d_out, int out_size, void* d_ws, size_t ws_size,
                              hipStream_t stream) {
    const float* inp = (const float*)d_in[0];   // [8,1,256,256] f32
    const int* tgt = (const int*)d_in[1];       // [8,1,256,256] i32
    float* out = (float*)d_out;                 // [1] f32

    float* g2 = (float*)d_ws;                   // 4*8*256*256 f32 = 8 MB
    float* partials = g2 + 4 * BB * HH * WW;    // 2048 f32

    edt_pass1<<<4 * BB, WW, 0, stream>>>(inp, tgt, g2);
    edt_pass2_loss<<<BB * HH, WW, 0, stream>>>(inp, tgt, g2, partials);
    wmma_reduce<<<1, 32, 0, stream>>>(partials, out, BB * HH,
                                      1.0f / (float)(BB * HH * WW));
}